// Loss_43353399886291
// MI455X (gfx1250) — compile-verified
//
#include <hip/hip_runtime.h>
#include <hip/hip_bf16.h>

#define BATCH      4
#define N_PATCHES  24
#define M_SAMP     256
#define NPTS       (N_PATCHES * M_SAMP)   /* 6144 points per batch */
#define MT         4096                   /* targets per batch */
#define GRID_PTS   64                     /* 8x8 grid per patch */
#define N_TRIS     98                     /* 2*7*7 */
#define N_ADJ      24
#define N_NONADJ   252
#define N_PAIRS    276
#define INV_SIGMA2 100.0f                 /* 1/sigma^2, sigma=0.1 */

typedef float v2f __attribute__((ext_vector_type(2)));
typedef float v8f __attribute__((ext_vector_type(8)));
typedef unsigned int u32x4 __attribute__((ext_vector_type(4)));
typedef int i32x4 __attribute__((ext_vector_type(4)));
typedef int i32x8 __attribute__((ext_vector_type(8)));

/* ---------------- workspace layout (bytes) ---------------- */
#define ACC_OFF      0                                            /* 8 floats  */
#define COLPACK_OFF  256                                          /* B*MT u64  */
#define CA_OFF       (COLPACK_OFF + BATCH*MT*8)                   /* B*N f32   */
#define IA_OFF       (CA_OFF + BATCH*NPTS*4)                      /* B*N i32   */
#define GP_OFF       (IA_OFF + BATCH*NPTS*4)                      /* B*P*64*3  */
#define MASK_OFF     (GP_OFF + BATCH*N_PATCHES*GRID_PTS*3*4)      /* 276 i32   */

/* chamfer kernel dynamic LDS: packed targets [MT][3] f32 + column table [MT] u64 */
#define CH_TGT_BYTES  (MT * 3 * 4)
#define CH_LDS_BYTES  (CH_TGT_BYTES + MT * 8)

/* acc[] indices: 0 chamfer, 1 normals, 2 collision, 3 planar, 4 template */

/* ---------------- reduction helpers (blockDim == 256) ---------------- */
__device__ __forceinline__ float blockReduceSum(float v, float* sred) {
    __syncthreads();
#pragma unroll
    for (int off = 16; off > 0; off >>= 1) v += __shfl_xor(v, off, 32);
    int wv = threadIdx.x >> 5;
    if ((threadIdx.x & 31) == 0) sred[wv] = v;
    __syncthreads();
    float r = 0.0f;
    if (threadIdx.x < 8) r = sred[threadIdx.x];
    if (threadIdx.x < 32) {
#pragma unroll
        for (int off = 4; off > 0; off >>= 1) r += __shfl_xor(r, off, 32);
    }
    if (threadIdx.x == 0) sred[0] = r;
    __syncthreads();
    return sred[0];
}

__device__ __forceinline__ float blockReduceMin(float v, float* sred) {
    __syncthreads();
#pragma unroll
    for (int off = 16; off > 0; off >>= 1) v = fminf(v, __shfl_xor(v, off, 32));
    int wv = threadIdx.x >> 5;
    if ((threadIdx.x & 31) == 0) sred[wv] = v;
    __syncthreads();
    float r = 3.0e38f;
    if (threadIdx.x < 8) r = sred[threadIdx.x];
    if (threadIdx.x < 32) {
#pragma unroll
        for (int off = 4; off > 0; off >>= 1) r = fminf(r, __shfl_xor(r, off, 32));
    }
    if (threadIdx.x == 0) sred[0] = r;
    __syncthreads();
    return sred[0];
}

/* order-preserving float->u32 key (unsigned compare == float compare) */
__device__ __forceinline__ unsigned fkey(float f) {
    unsigned u = __float_as_uint(f);
    return u ^ (unsigned)(((int)u >> 31) | 0x80000000);
}

/* ---------------- init: col-min table to +inf keys, accumulators to 0 ------ */
__global__ void k_init(unsigned long long* colpack, float* acc) {
    int t = blockIdx.x * blockDim.x + threadIdx.x;
    if (t < BATCH * MT) colpack[t] = 0xFFFFFFFFFFFFFFFFull;
    if (t < 8) acc[t] = 0.0f;
}

/* ---------------- Coons patch sampling on the 8x8 grid -------------------- */
__device__ __forceinline__ void bez(float u, float* w) {
    float v = 1.0f - u;
    w[0] = v * v * v;
    w[1] = 3.0f * u * v * v;
    w[2] = 3.0f * u * u * v;
    w[3] = u * u * u;
}

__global__ void k_coons(const float* __restrict__ patches, float* __restrict__ gp) {
    int bp = blockIdx.x;                 /* b*24 + p, 96 blocks */
    const float* cp = patches + (size_t)bp * 36;
    int m = threadIdx.x;                 /* 0..63 */
    int gi = m >> 3, gj = m & 7;
    float s = (float)gi * (1.0f / 7.0f);
    float t = (float)gj * (1.0f / 7.0f);
    float bs[4], bt[4], bs1[4], bt1[4];
    bez(s, bs); bez(t, bt); bez(1.0f - s, bs1); bez(1.0f - t, bt1);
    const int idx3[4] = {9, 10, 11, 0};
#pragma unroll
    for (int c = 0; c < 3; c++) {
        float cbv = 0.f, crv = 0.f, ctv = 0.f, clv = 0.f;
#pragma unroll
        for (int k = 0; k < 4; k++) {
            cbv += bs[k]  * cp[k * 3 + c];
            crv += bt[k]  * cp[(3 + k) * 3 + c];
            ctv += bs1[k] * cp[(6 + k) * 3 + c];
            clv += bt1[k] * cp[idx3[k] * 3 + c];
        }
        float P00 = cp[0 + c], P10 = cp[9 + c], P11 = cp[18 + c], P01 = cp[27 + c];
        float ruled = (1.f - t) * cbv + t * ctv + (1.f - s) * clv + s * crv;
        float bil = (1.f - s) * (1.f - t) * P00 + s * (1.f - t) * P10 +
                    s * t * P11 + (1.f - s) * t * P01;
        gp[((size_t)bp * GRID_PTS + m) * 3 + c] = ruled - bil;
    }
}

/* ---------------- pair enumeration + bbox masks ---------------------------- */
__device__ __forceinline__ void nonadjPair(int kn, int* pi, int* pj) {
    int cnt = 0;
    for (int i = 0; i < N_PATCHES; i++) {
        for (int j = i + 1; j < N_PATCHES; j++) {
            bool adj = (j == i + 1) || (i == 0 && j == N_PATCHES - 1);
            if (!adj) {
                if (cnt == kn) { *pi = i; *pj = j; return; }
                cnt++;
            }
        }
    }
    *pi = 0; *pj = 0;
}

__global__ void k_masks(const float* __restrict__ gp, int* __restrict__ masks) {
    int k = threadIdx.x;
    if (k >= N_PAIRS) return;
    int pa, pb;
    bool isadj = (k < N_ADJ);
    if (isadj) { pa = k; pb = (k + 1) % N_PATCHES; }
    else       { nonadjPair(k - N_ADJ, &pa, &pb); }
    int any = 0;
    for (int b = 0; b < BATCH; b++) {
        float mn1[3] = {3e38f, 3e38f, 3e38f}, mx1[3] = {-3e38f, -3e38f, -3e38f};
        float mn2[3] = {3e38f, 3e38f, 3e38f}, mx2[3] = {-3e38f, -3e38f, -3e38f};
        const float* g1 = gp + (size_t)(b * N_PATCHES + pa) * GRID_PTS * 3;
        const float* g2 = gp + (size_t)(b * N_PATCHES + pb) * GRID_PTS * 3;
        if (isadj) {
            for (int q = 0; q < 8; q++) {
                const float* v = g1 + (56 + q) * 3;
                for (int c = 0; c < 3; c++) { mn1[c] = fminf(mn1[c], v[c]); mx1[c] = fmaxf(mx1[c], v[c]); }
            }
        } else {
            for (int q = 0; q < GRID_PTS; q++) {
                const float* v = g1 + q * 3;
                for (int c = 0; c < 3; c++) { mn1[c] = fminf(mn1[c], v[c]); mx1[c] = fmaxf(mx1[c], v[c]); }
            }
        }
        for (int q = 0; q < GRID_PTS; q++) {
            const float* v = g2 + q * 3;
            for (int c = 0; c < 3; c++) { mn2[c] = fminf(mn2[c], v[c]); mx2[c] = fmaxf(mx2[c], v[c]); }
        }
        bool inter = true;
        for (int c = 0; c < 3; c++) inter = inter && (mx1[c] >= mn2[c]) && (mx2[c] >= mn1[c]);
        any |= (int)inter;
    }
    masks[k] = any;
}

/* ---------------- chamfer via V_WMMA_F32_16X16X4_F32 -----------------------
 * One wave = 16 points, sweeping 256 column tiles of 16 targets.
 * A = -2*p (16x4, K padded with 0), B = t (4x16), C = |p|^2 + |t|^2
 *   -> D tile = squared distances, directly from the WMMA.
 * Targets (xyz, stride-6 source) staged into LDS packed [MT][3] by the
 * Tensor Data Mover (2D D#: tile_dim0=3 elems, tile_dim1=MT rows, stride 6).
 * Row min/argmin: branchless selects in registers + shfl butterfly.
 * Col min/argmin: packed (float-key<<32 | row) u64, ds_min_u64 into an LDS
 * table, drained once per block with a global atomicMin.
 */
__global__ void __launch_bounds__(256)
k_chamfer(const float* __restrict__ points, const float* __restrict__ target,
          float* __restrict__ ca, int* __restrict__ ia,
          unsigned long long* __restrict__ colpack) {
    extern __shared__ char smem[];
    float* tgt = (float*)smem;                                       /* [MT][3] */
    unsigned long long* colTab = (unsigned long long*)(smem + CH_TGT_BYTES);

    const int blocksPerBatch = NPTS / 128;           /* 48 */
    int batch = blockIdx.x / blocksPerBatch;
    int blk   = blockIdx.x % blocksPerBatch;
    int tid   = threadIdx.x;
    int wave  = tid >> 5;
    int lane  = tid & 31;
    int i0    = blk * 128 + wave * 16;
    int halfBase = (lane & 16) ? 8 : 0;
    int col   = lane & 15;
    int rowIdxBase = i0 + halfBase;

    /* init per-block LDS column-min table */
    for (int m = tid; m < MT; m += 256) colTab[m] = 0xFFFFFFFFFFFFFFFFull;

    /* TDM stage: gather xyz (3 of 6 floats per row) for all MT targets */
    if (tid == 0) {
        unsigned long long gaddr =
            (unsigned long long)(const void*)(target + (size_t)batch * MT * 6);
        unsigned ldsOff = (unsigned)(unsigned long long)(void*)tgt;
        u32x4 g0;
        g0[0] = 1u;                                          /* count=1, user D# */
        g0[1] = ldsOff;                                      /* lds_addr (bytes) */
        g0[2] = (unsigned)(gaddr & 0xFFFFFFFFull);           /* global_addr lo   */
        g0[3] = (unsigned)((gaddr >> 32) & 0x01FFFFFFull)    /* global_addr hi   */
              | (2u << 30);                                  /* type = 2 (image) */
        i32x8 g1;
        g1[0] = (int)(2u << 16);          /* data_size=2 (4B), no pad/iter/mask  */
        g1[1] = (int)(6u << 16);          /* tensor_dim0 = 6 elems per row       */
        g1[2] = (int)((unsigned)MT << 16);/* tensor_dim1 = MT rows (lo16)        */
        g1[3] = (int)(3u << 16);          /* tensor_dim1 hi=0 | tile_dim0 = 3    */
        g1[4] = (int)MT;                  /* tile_dim1 = MT | tile_dim2 = 0      */
        g1[5] = 6;                        /* tensor_dim0_stride = 6 elems        */
        g1[6] = 0;
        g1[7] = 0;
        i32x4 gz4;
        gz4[0] = 0; gz4[1] = 0; gz4[2] = 0; gz4[3] = 0;
        i32x8 gz8;
        gz8[0] = 0; gz8[1] = 0; gz8[2] = 0; gz8[3] = 0;
        gz8[4] = 0; gz8[5] = 0; gz8[6] = 0; gz8[7] = 0;
        __builtin_amdgcn_tensor_load_to_lds(g0, g1, gz4, gz4, gz8, 0);
        __builtin_amdgcn_s_wait_tensorcnt(0);
    }
    __syncthreads();

    /* A tile: 16 points scaled by -2, K=3 padded to 4 */
    const float* pp = points + ((size_t)batch * NPTS + i0 + col) * 3;
    float px = pp[0], py = pp[1], pz = pp[2];
    float pn2 = px * px + py * py + pz * pz;
    v2f amat;
    if (lane < 16) { amat.x = -2.0f * px; amat.y = -2.0f * py; }
    else           { amat.x = -2.0f * pz; amat.y = 0.0f; }

    float pn2row[8];
#pragma unroll
    for (int r = 0; r < 8; r++) pn2row[r] = __shfl(pn2, halfBase + r, 32);

    float rminv[8];
    int   rargv[8];
#pragma unroll
    for (int r = 0; r < 8; r++) { rminv[r] = 3.0e38f; rargv[r] = 0; }

#pragma unroll 4
    for (int j0 = 0; j0 < MT; j0 += 16) {
        const float* tt = tgt + (size_t)(j0 + col) * 3;
        float tx = tt[0], ty = tt[1], tz = tt[2];
        float tn2 = tx * tx + ty * ty + tz * tz;
        v2f bmat;
        if (lane < 16) { bmat.x = tx; bmat.y = ty; }
        else           { bmat.x = tz; bmat.y = 0.0f; }
        v8f cmat;
#pragma unroll
        for (int r = 0; r < 8; r++) cmat[r] = pn2row[r] + tn2;

        v8f d = __builtin_amdgcn_wmma_f32_16x16x4_f32(
            false, amat, false, bmat, (short)0, cmat, false, false);

        unsigned long long colBest = 0xFFFFFFFFFFFFFFFFull;
#pragma unroll
        for (int r = 0; r < 8; r++) {
            float dv = d[r];
            /* row running min/argmin: branchless */
            bool lt = dv < rminv[r];
            rminv[r] = lt ? dv : rminv[r];
            rargv[r] = lt ? (j0 + col) : rargv[r];
            /* column candidate: (key, row) packed -> unsigned min == argmin */
            unsigned long long pk =
                ((unsigned long long)fkey(dv) << 32) | (unsigned)(rowIdxBase + r);
            colBest = (pk < colBest) ? pk : colBest;
        }
        /* combine the two 8-row halves */
        unsigned hi = (unsigned)(colBest >> 32);
        unsigned lo = (unsigned)colBest;
        unsigned ohi = (unsigned)__shfl_xor((int)hi, 16, 32);
        unsigned olo = (unsigned)__shfl_xor((int)lo, 16, 32);
        unsigned long long other = ((unsigned long long)ohi << 32) | olo;
        colBest = (other < colBest) ? other : colBest;
        if (lane < 16) atomicMin(&colTab[j0 + lane], colBest);   /* ds_min_u64 */
    }

    /* reduce row min/argmin across the 16 lanes of each half */
#pragma unroll
    for (int off = 1; off < 16; off <<= 1) {
#pragma unroll
        for (int r = 0; r < 8; r++) {
            float ov = __shfl_xor(rminv[r], off, 32);
            int   oa = __shfl_xor(rargv[r], off, 32);
            bool lt = ov < rminv[r];
            rminv[r] = lt ? ov : rminv[r];
            rargv[r] = lt ? oa : rargv[r];
        }
    }
    if (col == 0) {
#pragma unroll
        for (int r = 0; r < 8; r++) {
            ca[(size_t)batch * NPTS + rowIdxBase + r] = rminv[r];
            ia[(size_t)batch * NPTS + rowIdxBase + r] = rargv[r];
        }
    }

    /* drain per-block column table into the global one */
    __syncthreads();
    for (int m = tid; m < MT; m += 256)
        atomicMin(&colpack[(size_t)batch * MT + m], colTab[m]);
}

/* ---------------- chamfer / normals finalize ------------------------------- */
__global__ void k_chamfer_fin(const float* __restrict__ normals, const float* __restrict__ target,
                              const float* __restrict__ mtds,
                              const float* __restrict__ ca, const int* __restrict__ ia,
                              const unsigned long long* __restrict__ colpack,
                              float* __restrict__ acc) {
    __shared__ float sred[8];
    int b = blockIdx.x;
    int tid = threadIdx.x;
    float sw = 0.f, swc = 0.f, swn = 0.f;
    for (int n = tid; n < NPTS; n += 256) {
        size_t off = (size_t)b * NPTS + n;
        float w = mtds[off];
        sw  += w;
        swc += w * ca[off];
        int iam = ia[off];
        const float* nr = normals + off * 3;
        const float* tn = target + ((size_t)b * MT + iam) * 6 + 3;
        float dx = nr[0] - tn[0], dy = nr[1] - tn[1], dz = nr[2] - tn[2];
        swn += w * (dx * dx + dy * dy + dz * dz);
    }
    sw  = blockReduceSum(sw, sred);
    swc = blockReduceSum(swc, sred);
    swn = blockReduceSum(swn, sred);

    float scb = 0.f, snb = 0.f;
    for (int m = tid; m < MT; m += 256) {
        unsigned long long pk = colpack[(size_t)b * MT + m];
        unsigned key = (unsigned)(pk >> 32);
        unsigned u = (key & 0x80000000u) ? (key & 0x7FFFFFFFu) : ~key;
        float cbv = __uint_as_float(u);
        int ib = (int)(pk & 0xFFFFFFFFull);
        scb += cbv;
        const float* tn = target + ((size_t)b * MT + m) * 6 + 3;
        const float* nr = normals + ((size_t)b * NPTS + ib) * 3;
        float dx = tn[0] - nr[0], dy = tn[1] - nr[1], dz = tn[2] - nr[2];
        snb += dx * dx + dy * dy + dz * dz;
    }
    scb = blockReduceSum(scb, sred);
    snb = blockReduceSum(snb, sred);

    if (tid == 0) {
        atomicAdd(&acc[0], (swc / sw + scb / (float)MT) * 0.5f * 0.25f);
        atomicAdd(&acc[1], (swn / sw + snb / (float)MT) * 0.5f * 0.25f);
    }
}

/* ---------------- template normals loss ------------------------------------ */
__global__ void k_template(const float* __restrict__ normals,
                           const float* __restrict__ template_normals,
                           const float* __restrict__ mtds, float* __restrict__ acc) {
    __shared__ float sred[8];
    int bp = blockIdx.x;            /* b*24+p, 96 blocks */
    int p  = bp % N_PATCHES;
    int m  = threadIdx.x;           /* 0..255 */
    size_t off = (size_t)bp * M_SAMP + m;
    float w = mtds[off];
    const float* nr = normals + off * 3;
    const float* tn = template_normals + ((size_t)p * M_SAMP + m) * 3;
    float dx = tn[0] - nr[0], dy = tn[1] - nr[1], dz = tn[2] - nr[2];
    float l = dx * dx + dy * dy + dz * dz;
    float sw  = blockReduceSum(w, sred);
    float swl = blockReduceSum(w * l, sred);
    if (threadIdx.x == 0) atomicAdd(&acc[4], (swl / sw) * (1.0f / 96.0f));
}

/* ---------------- planar patch loss ---------------------------------------- */
__global__ void k_planar(const float* __restrict__ points, const float* __restrict__ st,
                         const float* __restrict__ mtds, float* __restrict__ acc) {
    __shared__ float sred[8];
    __shared__ float ssol[9];
    int bp = blockIdx.x;            /* 96 blocks */
    int m  = threadIdx.x;           /* 0..255 */
    float s = st[m * 2 + 0];
    float t = st[m * 2 + 1];
    const float* pt = points + ((size_t)bp * M_SAMP + m) * 3;
    float p0 = pt[0], p1 = pt[1], p2 = pt[2];

    float a00 = blockReduceSum(s * s, sred);
    float a01 = blockReduceSum(s * t, sred);
    float a02 = blockReduceSum(s, sred);
    float a11 = blockReduceSum(t * t, sred);
    float a12 = blockReduceSum(t, sred);
    float a22 = (float)M_SAMP;

    float atp[9];
    atp[0] = blockReduceSum(s * p0, sred);
    atp[1] = blockReduceSum(s * p1, sred);
    atp[2] = blockReduceSum(s * p2, sred);
    atp[3] = blockReduceSum(t * p0, sred);
    atp[4] = blockReduceSum(t * p1, sred);
    atp[5] = blockReduceSum(t * p2, sred);
    atp[6] = blockReduceSum(p0, sred);
    atp[7] = blockReduceSum(p1, sred);
    atp[8] = blockReduceSum(p2, sred);

    if (threadIdx.x == 0) {
        float det = a00 * (a11 * a22 - a12 * a12)
                  - a01 * (a01 * a22 - a12 * a02)
                  + a02 * (a01 * a12 - a11 * a02);
        float id = 1.0f / det;
        float i00 = (a11 * a22 - a12 * a12) * id;
        float i01 = (a02 * a12 - a01 * a22) * id;
        float i02 = (a01 * a12 - a02 * a11) * id;
        float i11 = (a00 * a22 - a02 * a02) * id;
        float i12 = (a02 * a01 - a00 * a12) * id;
        float i22 = (a00 * a11 - a01 * a01) * id;
        for (int c = 0; c < 3; c++) {
            ssol[0 * 3 + c] = i00 * atp[0 + c] + i01 * atp[3 + c] + i02 * atp[6 + c];
            ssol[1 * 3 + c] = i01 * atp[0 + c] + i11 * atp[3 + c] + i12 * atp[6 + c];
            ssol[2 * 3 + c] = i02 * atp[0 + c] + i12 * atp[3 + c] + i22 * atp[6 + c];
        }
    }
    __syncthreads();
    float rx = p0 - (s * ssol[0] + t * ssol[3] + ssol[6]);
    float ry = p1 - (s * ssol[1] + t * ssol[4] + ssol[7]);
    float rz = p2 - (s * ssol[2] + t * ssol[5] + ssol[8]);
    float l = rx * rx + ry * ry + rz * rz;
    float w = mtds[(size_t)bp * M_SAMP + m];
    float sw  = blockReduceSum(w, sred);
    float swl = blockReduceSum(w * l, sred);
    if (threadIdx.x == 0) atomicAdd(&acc[3], (swl / sw) * (1.0f / 96.0f));
}

/* ---------------- point-triangle distance (faithful where-chain) ----------- */
__device__ __forceinline__ float safe12(float x) {
    return (fabsf(x) < 1e-12f) ? 1e-12f : x;
}

__device__ float ptTriSqDist(const float* p, const float* a, const float* b, const float* c) {
    float ab[3], ac[3], ap[3], bp[3], cpv[3];
#pragma unroll
    for (int i = 0; i < 3; i++) {
        ab[i] = b[i] - a[i]; ac[i] = c[i] - a[i];
        ap[i] = p[i] - a[i]; bp[i] = p[i] - b[i]; cpv[i] = p[i] - c[i];
    }
    float d1 = ab[0] * ap[0] + ab[1] * ap[1] + ab[2] * ap[2];
    float d2 = ac[0] * ap[0] + ac[1] * ap[1] + ac[2] * ap[2];
    float d3 = ab[0] * bp[0] + ab[1] * bp[1] + ab[2] * bp[2];
    float d4 = ac[0] * bp[0] + ac[1] * bp[1] + ac[2] * bp[2];
    float d5 = ab[0] * cpv[0] + ab[1] * cpv[1] + ab[2] * cpv[2];
    float d6 = ac[0] * cpv[0] + ac[1] * cpv[1] + ac[2] * cpv[2];
    float vc = d1 * d4 - d3 * d2;
    float vb = d5 * d2 - d1 * d6;
    float va = d3 * d6 - d5 * d4;
    float t_ab = fminf(fmaxf(d1 / safe12(d1 - d3), 0.f), 1.f);
    float t_ac = fminf(fmaxf(d2 / safe12(d2 - d6), 0.f), 1.f);
    float t_bc = fminf(fmaxf((d4 - d3) / safe12((d4 - d3) + (d5 - d6)), 0.f), 1.f);
    float denom = safe12(va + vb + vc);
    float v = vb / denom, w = vc / denom;
    float close[3];
#pragma unroll
    for (int i = 0; i < 3; i++) close[i] = a[i] + ab[i] * v + ac[i] * w;
    if (va <= 0.f && (d4 - d3) >= 0.f && (d5 - d6) >= 0.f)
        for (int i = 0; i < 3; i++) close[i] = b[i] + (c[i] - b[i]) * t_bc;
    if (vb <= 0.f && d2 >= 0.f && d6 <= 0.f)
        for (int i = 0; i < 3; i++) close[i] = a[i] + ac[i] * t_ac;
    if (d6 >= 0.f && d5 <= d6)
        for (int i = 0; i < 3; i++) close[i] = c[i];
    if (vc <= 0.f && d1 >= 0.f && d3 <= 0.f)
        for (int i = 0; i < 3; i++) close[i] = a[i] + ab[i] * t_ab;
    if (d3 >= 0.f && d4 <= d3)
        for (int i = 0; i < 3; i++) close[i] = b[i];
    if (d1 <= 0.f && d2 <= 0.f)
        for (int i = 0; i < 3; i++) close[i] = a[i];
    float dx = p[0] - close[0], dy = p[1] - close[1], dz = p[2] - close[2];
    return dx * dx + dy * dy + dz * dz;
}

__device__ __forceinline__ void triVerts(int t, int* v0, int* v1, int* v2) {
    int cell = t >> 1;
    int ci = cell / 7, cj = cell % 7;
    int v00 = ci * 8 + cj;
    if ((t & 1) == 0) { *v0 = v00; *v1 = v00 + 8; *v2 = v00 + 1; }
    else              { *v0 = v00 + 8; *v1 = v00 + 9; *v2 = v00 + 1; }
}

/* ---------------- collision loss ------------------------------------------- */
__global__ void k_collision(const float* __restrict__ gp, const int* __restrict__ masks,
                            float* __restrict__ acc) {
    __shared__ float sred[8];
    int bid = blockIdx.x;           /* BATCH * 276 blocks */
    int b = bid / N_PAIRS;
    int k = bid % N_PAIRS;
    bool isadj = (k < N_ADJ);
    int pa, pb;
    if (isadj) { pa = k; pb = (k + 1) % N_PATCHES; }
    else       { nonadjPair(k - N_ADJ, &pa, &pb); }
    const float* g1 = gp + (size_t)(b * N_PATCHES + pa) * GRID_PTS * 3;
    const float* g2 = gp + (size_t)(b * N_PATCHES + pb) * GRID_PTS * 3;

    int total = isadj ? (8 * N_TRIS + GRID_PTS * 7) : (GRID_PTS * N_TRIS * 2);
    float mind = 3.0e38f;
    for (int cc = threadIdx.x; cc < total; cc += 256) {
        const float *P, *gt;
        int tri;
        if (isadj) {
            if (cc < 8 * N_TRIS) { P = g1 + (56 + cc / N_TRIS) * 3; tri = cc % N_TRIS; gt = g2; }
            else { int c2 = cc - 8 * N_TRIS; P = g2 + (c2 / 7) * 3; tri = 85 + 2 * (c2 % 7); gt = g1; }
        } else {
            if (cc < GRID_PTS * N_TRIS) { P = g1 + (cc / N_TRIS) * 3; tri = cc % N_TRIS; gt = g2; }
            else { int c2 = cc - GRID_PTS * N_TRIS; P = g2 + (c2 / N_TRIS) * 3; tri = c2 % N_TRIS; gt = g1; }
        }
        int v0, v1, v2;
        triVerts(tri, &v0, &v1, &v2);
        float d2v = ptTriSqDist(P, gt + v0 * 3, gt + v1 * 3, gt + v2 * 3);
        mind = fminf(mind, d2v);
    }
    mind = blockReduceMin(mind, sred);
    if (threadIdx.x == 0) {
        float contrib = masks[k] ? expf(-(mind + 1e-12f) * INV_SIGMA2) : 0.0f;
        atomicAdd(&acc[2], contrib * 0.25f);   /* mean over batch */
    }
}

/* ---------------- combine -------------------------------------------------- */
__global__ void k_combine(const float* __restrict__ acc, float* __restrict__ out) {
    if (threadIdx.x == 0 && blockIdx.x == 0) {
        float ch = acc[0], no = acc[1], co = acc[2], pl = acc[3], te = acc[4];
        out[0] = ch + no + 0.01f * co + pl + te;   /* + 0 * symmetry */
        out[1] = ch;
        out[2] = no;
        out[3] = co;
        out[4] = pl;
        out[5] = te;
        out[6] = 0.0f;
    }
}

/* ---------------- launcher ------------------------------------------------- */
extern "C" void kernel_launch(void* const* d_in, const int* in_sizes, int n_in,
                              void* d_out, int out_size, void* d_ws, size_t ws_size,
                              hipStream_t stream) {
    (void)in_sizes; (void)n_in; (void)out_size; (void)ws_size;
    const float* target           = (const float*)d_in[0];
    const float* patches          = (const float*)d_in[1];
    const float* points           = (const float*)d_in[2];
    const float* normals          = (const float*)d_in[3];
    const float* mtds             = (const float*)d_in[4];
    const float* st               = (const float*)d_in[5];
    const float* template_normals = (const float*)d_in[6];
    float* out = (float*)d_out;

    char* ws = (char*)d_ws;
    float*              acc     = (float*)(ws + ACC_OFF);
    unsigned long long* colpack = (unsigned long long*)(ws + COLPACK_OFF);
    float*              ca      = (float*)(ws + CA_OFF);
    int*                ia      = (int*)(ws + IA_OFF);
    float*              gp      = (float*)(ws + GP_OFF);
    int*                masks   = (int*)(ws + MASK_OFF);

    k_init<<<64, 256, 0, stream>>>(colpack, acc);
    k_coons<<<BATCH * N_PATCHES, GRID_PTS, 0, stream>>>(patches, gp);
    k_masks<<<1, 288, 0, stream>>>(gp, masks);
    k_chamfer<<<BATCH * (NPTS / 128), 256, CH_LDS_BYTES, stream>>>(points, target, ca, ia, colpack);
    k_chamfer_fin<<<BATCH, 256, 0, stream>>>(normals, target, mtds, ca, ia, colpack, acc);
    k_template<<<BATCH * N_PATCHES, 256, 0, stream>>>(normals, template_normals, mtds, acc);
    k_planar<<<BATCH * N_PATCHES, 256, 0, stream>>>(points, st, mtds, acc);
    k_collision<<<BATCH * N_PAIRS, 256, 0, stream>>>(gp, masks, acc);
    k_combine<<<1, 64, 0, stream>>>(acc, out);
}